// GCN_57440892616779
// MI455X (gfx1250) — compile-verified
//
#include <hip/hip_runtime.h>
#include <math.h>

typedef float v2f __attribute__((ext_vector_type(2)));
typedef float v8f __attribute__((ext_vector_type(8)));

constexpr int F_IN = 512;
constexpr int HID  = 128;
constexpr int NCLS = 64;

// ---------------- degree / normalization ----------------

__global__ void k_init_deg(float* __restrict__ deg, int n) {
    int i = blockIdx.x * blockDim.x + threadIdx.x;
    if (i < n) deg[i] = 1.0f;  // implicit unit self-loop
}

__global__ void k_deg_scatter(const int* __restrict__ row,
                              const float* __restrict__ ew,
                              float* __restrict__ deg, int E) {
    int e = blockIdx.x * blockDim.x + threadIdx.x;
    if (e < E) unsafeAtomicAdd(&deg[row[e]], ew[e]);
}

__global__ void k_dinv(float* __restrict__ deg, int n) {
    int i = blockIdx.x * blockDim.x + threadIdx.x;
    if (i < n) {
        float d = deg[i];
        deg[i] = (d > 0.0f) ? rsqrtf(d) : 0.0f;
    }
}

// ---------------- CSR build (atomic-free aggregation prep) ----------------

__global__ void k_zero_i32(int* __restrict__ p, int n) {
    int i = blockIdx.x * blockDim.x + threadIdx.x;
    if (i < n) p[i] = 0;
}

__global__ void k_count(const int* __restrict__ row, int* __restrict__ cnt, int E) {
    int e = blockIdx.x * blockDim.x + threadIdx.x;
    if (e < E) atomicAdd(&cnt[row[e]], 1);
}

// Block-level exclusive scan: 256 threads x 4 elements = 1024 per block.
__global__ __launch_bounds__(256) void k_scan1(const int* __restrict__ cnt,
                                               int* __restrict__ out,
                                               int* __restrict__ bsum, int n) {
    __shared__ int wsum[8];
    const int t = threadIdx.x;
    const int lane = t & 31, w = t >> 5;
    const int base = blockIdx.x * 1024 + t * 4;
    int v[4], tsum = 0;
#pragma unroll
    for (int j = 0; j < 4; ++j) {
        v[j] = (base + j < n) ? cnt[base + j] : 0;
        tsum += v[j];
    }
    // inclusive wave scan of per-thread sums
    int sc = tsum;
    for (int off = 1; off < 32; off <<= 1) {
        int u = __shfl_up(sc, off, 32);
        if (lane >= off) sc += u;
    }
    if (lane == 31) wsum[w] = sc;
    __syncthreads();
    int woff = 0;
    for (int k = 0; k < w; ++k) woff += wsum[k];
    int run = woff + sc - tsum;  // exclusive prefix for this thread
#pragma unroll
    for (int j = 0; j < 4; ++j) {
        if (base + j < n) out[base + j] = run;
        run += v[j];
    }
    if (t == 255) bsum[blockIdx.x] = woff + sc;  // block total
}

__global__ void k_scan2(int* __restrict__ bsum, int nb) {
    if (blockIdx.x == 0 && threadIdx.x == 0) {
        int acc = 0;
        for (int i = 0; i < nb; ++i) { int t = bsum[i]; bsum[i] = acc; acc += t; }
    }
}

__global__ void k_scan3(int* __restrict__ rowptr, const int* __restrict__ bsum,
                        int n, int E) {
    int i = blockIdx.x * blockDim.x + threadIdx.x;
    if (i < n) rowptr[i] += bsum[i >> 10];
    if (i == 0) rowptr[n] = E;
}

// bucket-fill: sorted (col, norm) per target node; norm computed once per edge
__global__ void k_fill(const int* __restrict__ rowi, const int* __restrict__ coli,
                       const float* __restrict__ ew, const float* __restrict__ dinv,
                       const int* __restrict__ rowptr, int* __restrict__ cursor,
                       int* __restrict__ scol, float* __restrict__ snorm, int E) {
    int e = blockIdx.x * blockDim.x + threadIdx.x;
    if (e >= E) return;
    int r = rowi[e], c = coli[e];
    int p = rowptr[r] + atomicAdd(&cursor[r], 1);
    scol[p] = c;
    snorm[p] = dinv[r] * ew[e] * dinv[c];
}

// ---------------- fp32 WMMA GEMM: C[M,NDIM] = A[M,KDIM] @ B[KDIM,NDIM] ----------------

template<int KDIM, int NDIM>
__global__ __launch_bounds__(256) void k_gemm_wmma(const float* __restrict__ A,
                                                   const float* __restrict__ B,
                                                   float* __restrict__ C, int M) {
    constexpr int NT = NDIM / 16;
    constexpr int KT = 64;
    __shared__ float Bs[KT * NDIM];

    const int tid  = threadIdx.x;
    const int wave = tid >> 5;
    const int lane = tid & 31;
    const int lo   = lane & 15;
    const int hi   = lane >> 4;

    const int m0 = blockIdx.x * 128 + wave * 16;
    int rowA = m0 + lo;
    if (rowA > M - 1) rowA = M - 1;  // clamp: EXEC must stay all-ones for WMMA
    const float2* Arow = reinterpret_cast<const float2*>(A + (size_t)rowA * KDIM);

    v8f acc[NT] = {};

    for (int k0 = 0; k0 < KDIM; k0 += KT) {
        __syncthreads();
        for (int i = tid; i < KT * NDIM; i += 256)
            Bs[i] = B[(size_t)k0 * NDIM + i];
        __syncthreads();

        for (int kk = 0; kk < KT; kk += 4) {
            float2 a2 = Arow[((k0 + kk) >> 1) + hi];
            v2f a; a.x = a2.x; a.y = a2.y;
#pragma unroll
            for (int nt = 0; nt < NT; ++nt) {
                v2f b;
                b.x = Bs[(kk + 2 * hi) * NDIM + nt * 16 + lo];
                b.y = Bs[(kk + 2 * hi + 1) * NDIM + nt * 16 + lo];
                acc[nt] = __builtin_amdgcn_wmma_f32_16x16x4_f32(
                    false, a, false, b, (short)0, acc[nt], false, false);
            }
        }
    }

#pragma unroll
    for (int nt = 0; nt < NT; ++nt) {
#pragma unroll
        for (int r = 0; r < 8; ++r) {
            int row = m0 + r + hi * 8;
            if (row < M)
                C[(size_t)row * NDIM + nt * 16 + lo] = acc[nt][r];
        }
    }
}

// ---------------- atomic-free gather aggregation ----------------
// One wave per node. Lanes jointly fetch 32 edges' (col,norm), broadcast each via
// shfl, accumulate PER features per lane in registers. Fused self-loop + bias (+ReLU).

template<int D, int PER, bool RELU>
__global__ __launch_bounds__(256) void k_gather_agg(const float* __restrict__ h,
                                                    const float* __restrict__ dinv,
                                                    const int* __restrict__ rowptr,
                                                    const int* __restrict__ scol,
                                                    const float* __restrict__ snorm,
                                                    const float* __restrict__ bias,
                                                    float* __restrict__ out, int n) {
    int g = (int)(((size_t)blockIdx.x * blockDim.x + threadIdx.x) >> 5);
    int lane = threadIdx.x & 31;
    if (g >= n) return;

    float acc[PER];
    float s = dinv[g];
    float ss = s * s;
    const float* hrow = h + (size_t)g * D + lane * PER;
#pragma unroll
    for (int j = 0; j < PER; ++j) acc[j] = hrow[j] * ss;  // self-loop term

    int beg = rowptr[g], end = rowptr[g + 1];
    for (int p0 = beg; p0 < end; p0 += 32) {
        int pe = p0 + lane;
        int c = 0; float nm = 0.0f;
        if (pe < end) { c = scol[pe]; nm = snorm[pe]; }
        int cnt = end - p0; if (cnt > 32) cnt = 32;
        for (int j = 0; j < cnt; ++j) {
            int   cj = __shfl(c, j, 32);
            float nj = __shfl(nm, j, 32);
            const float* hc = h + (size_t)cj * D + lane * PER;
            if constexpr (PER == 4) {
                float4 v = *reinterpret_cast<const float4*>(hc);
                acc[0] += nj * v.x; acc[1] += nj * v.y;
                acc[2] += nj * v.z; acc[3] += nj * v.w;
            } else {
                float2 v = *reinterpret_cast<const float2*>(hc);
                acc[0] += nj * v.x; acc[1] += nj * v.y;
            }
        }
    }

    float* op = out + (size_t)g * D + lane * PER;
#pragma unroll
    for (int j = 0; j < PER; ++j) {
        float v = acc[j] + bias[lane * PER + j];
        if constexpr (RELU) v = v > 0.0f ? v : 0.0f;
        op[j] = v;
    }
}

// ---------------- log-softmax over 64 classes: one wave per node ----------------
// (bias already folded into layer-2 gather)

__global__ __launch_bounds__(256) void k_logsoftmax(const float* __restrict__ a,
                                                    float* __restrict__ out, int n) {
    int g = (int)(((size_t)blockIdx.x * blockDim.x + threadIdx.x) >> 5);
    int lane = threadIdx.x & 31;
    if (g >= n) return;
    float2 v = *reinterpret_cast<const float2*>(a + (size_t)g * NCLS + lane * 2);
    float x0 = v.x, x1 = v.y;
    float m = fmaxf(x0, x1);
    for (int off = 16; off > 0; off >>= 1)
        m = fmaxf(m, __shfl_xor(m, off, 32));
    float ssum = __expf(x0 - m) + __expf(x1 - m);
    for (int off = 16; off > 0; off >>= 1)
        ssum += __shfl_xor(ssum, off, 32);
    float l = __logf(ssum);
    float2 o; o.x = x0 - m - l; o.y = x1 - m - l;
    *reinterpret_cast<float2*>(out + (size_t)g * NCLS + lane * 2) = o;
}

// ---------------- launch ----------------

static inline size_t align256(size_t x) { return (x + 255) & ~(size_t)255; }

extern "C" void kernel_launch(void* const* d_in, const int* in_sizes, int n_in,
                              void* d_out, int out_size, void* d_ws, size_t ws_size,
                              hipStream_t stream) {
    const float* x  = (const float*)d_in[0];
    const float* W1 = (const float*)d_in[1];
    const float* b1 = (const float*)d_in[2];
    const float* W2 = (const float*)d_in[3];
    const float* b2 = (const float*)d_in[4];
    const float* ew = (const float*)d_in[5];
    const int*   ei = (const int*)d_in[6];

    const int Nn = in_sizes[0] / F_IN;
    const int E  = in_sizes[5];
    const int* rowi = ei;       // targets
    const int* coli = ei + E;   // sources

    // workspace carve-up
    char* ws = (char*)d_ws;
    size_t off = 0;
    float* dinv   = (float*)(ws + off); off += align256((size_t)Nn * 4);
    int*   rowptr = (int*)  (ws + off); off += align256((size_t)(Nn + 1) * 4);
    int*   cnt    = (int*)  (ws + off); off += align256((size_t)Nn * 4);   // also cursor
    int*   bsum   = (int*)  (ws + off); off += align256(4096);
    int*   scol   = (int*)  (ws + off); off += align256((size_t)E * 4);
    float* snorm  = (float*)(ws + off); off += align256((size_t)E * 4);
    float* h      = (float*)(ws + off); off += align256((size_t)Nn * HID * 4);
    float* hx     = (float*)(ws + off); off += align256((size_t)Nn * HID * 4);
    float* g2     = h;   // reuse: h dead after layer-1 gather
    float* agg2   = hx;  // reuse: h1 dead after GEMM2
    float* out    = (float*)d_out;

    const int TB = 256;
    const int nbN = (Nn + TB - 1) / TB;
    const int nbE = (E + TB - 1) / TB;
    const int nbScan = (Nn + 1023) / 1024;

    // degree -> dinv
    k_init_deg<<<nbN, TB, 0, stream>>>(dinv, Nn);
    k_deg_scatter<<<nbE, TB, 0, stream>>>(rowi, ew, dinv, E);
    k_dinv<<<nbN, TB, 0, stream>>>(dinv, Nn);

    // CSR build
    k_zero_i32<<<nbN, TB, 0, stream>>>(cnt, Nn);
    k_count<<<nbE, TB, 0, stream>>>(rowi, cnt, E);
    k_scan1<<<nbScan, TB, 0, stream>>>(cnt, rowptr, bsum, Nn);
    k_scan2<<<1, 32, 0, stream>>>(bsum, nbScan);
    k_scan3<<<nbN, TB, 0, stream>>>(rowptr, bsum, Nn, E);
    k_zero_i32<<<nbN, TB, 0, stream>>>(cnt, Nn);  // cursor
    k_fill<<<nbE, TB, 0, stream>>>(rowi, coli, ew, dinv, rowptr, cnt, scol, snorm, E);

    // layer 1: GEMM (WMMA f32) + gather aggregation (+b1, ReLU)
    k_gemm_wmma<F_IN, HID><<<(Nn + 127) / 128, TB, 0, stream>>>(x, W1, h, Nn);
    k_gather_agg<HID, 4, true><<<(Nn + 7) / 8, TB, 0, stream>>>(
        h, dinv, rowptr, scol, snorm, b1, hx, Nn);

    // layer 2: GEMM (WMMA f32) + gather aggregation (+b2)
    k_gemm_wmma<HID, NCLS><<<(Nn + 127) / 128, TB, 0, stream>>>(hx, W2, g2, Nn);
    k_gather_agg<NCLS, 2, false><<<(Nn + 7) / 8, TB, 0, stream>>>(
        g2, dinv, rowptr, scol, snorm, b2, agg2, Nn);

    // log-softmax
    k_logsoftmax<<<(Nn + 7) / 8, TB, 0, stream>>>(agg2, out, Nn);
}